// MeanPooling_8950711845799
// MI455X (gfx1250) — compile-verified
//
#include <hip/hip_runtime.h>

// Scatter-mean over segments:
//   out[s, :] = sum_{r : index[r]==s} x[r, :] / max(count(s), 1)
// x: [N=4e6, D=32] f32, index: [N] int, out: [S=1e5, D=32] f32.
//
// Memory-bound: 512 MB stream of x + 16 MB index + 12.8 MB out at 23.3 TB/s
// => ~24 us floor. Accumulators (12.8 MB sums + 0.4 MB counts) stay resident
// in the 192 MB L2 because x/index are loaded non-temporally; the 128 M f32
// scatter adds are forced to native non-returning global_atomic_add_f32
// (device scope -> resolved at L2) via inline asm so clang's fp-atomic
// CAS-loop fallback can never kick in.

typedef float v4f __attribute__((ext_vector_type(4)));

#define TPB 256

__device__ __forceinline__ void hw_atomic_add_f32(float* p, float v) {
    // Non-returning f32 atomic add, device scope (L2-resolved, all-WGP coherent).
    asm volatile("global_atomic_add_f32 %0, %1, off scope:SCOPE_DEV"
                 :: "v"(p), "v"(v)
                 : "memory");
}

__global__ void __launch_bounds__(TPB)
zero_init_kernel(float* __restrict__ out, unsigned* __restrict__ cnt,
                 int out_n, int seg_n) {
    const int i = blockIdx.x * blockDim.x + threadIdx.x;
    if (i < out_n) out[i] = 0.0f;
    if (i < seg_n) cnt[i] = 0u;
}

// One lane per eighth-row (4 floats). 8 lanes per row, 4 rows per wave32:
// each wave issues one fully-coalesced 512B global_load_b128 burst.
__global__ void __launch_bounds__(TPB)
scatter_add_kernel(const float* __restrict__ x,
                   const int* __restrict__ index,
                   float* __restrict__ sums,
                   unsigned* __restrict__ cnt,
                   int nquads, int seg_n) {
    const int q = blockIdx.x * blockDim.x + threadIdx.x;
    if (q >= nquads) return;

    const int row  = q >> 3;   // D=32 -> 8 quads per row
    const int part = q & 7;    // which 4-float chunk of the row

    // Streaming (non-temporal) loads: x is touched exactly once; keep it out
    // of L2 so the atomic accumulators stay resident there.
    const v4f v   = __builtin_nontemporal_load((const v4f*)x + q);
    const int seg = __builtin_nontemporal_load(index + row);
    if ((unsigned)seg >= (unsigned)seg_n) return;  // safety guard

    float* base = sums + (size_t)seg * 32 + (size_t)part * 4;
    hw_atomic_add_f32(base + 0, v.x);
    hw_atomic_add_f32(base + 1, v.y);
    hw_atomic_add_f32(base + 2, v.z);
    hw_atomic_add_f32(base + 3, v.w);

    if (part == 0) {
        // Integer atomics always lower natively (global_atomic_add_u32).
        __hip_atomic_fetch_add(cnt + seg, 1u, __ATOMIC_RELAXED,
                               __HIP_MEMORY_SCOPE_AGENT);
    }
}

__global__ void __launch_bounds__(TPB)
finalize_kernel(float* __restrict__ out, const unsigned* __restrict__ cnt,
                int out_n) {
    const int i = blockIdx.x * blockDim.x + threadIdx.x;
    if (i >= out_n) return;
    const unsigned c = cnt[i >> 5];                 // D=32 features per segment
    const float denom = (c > 0u) ? (float)c : 1.0f; // max(count, 1)
    out[i] = out[i] / denom;
}

extern "C" void kernel_launch(void* const* d_in, const int* in_sizes, int n_in,
                              void* d_out, int out_size, void* d_ws, size_t ws_size,
                              hipStream_t stream) {
    const float* x     = (const float*)d_in[0];   // [N, 32] f32
    const int*   index = (const int*)d_in[1];     // [N] int
    float*       out   = (float*)d_out;           // [S, 32] f32
    unsigned*    cnt   = (unsigned*)d_ws;         // [S] u32 counts (scratch)

    const int rows   = in_sizes[1];       // N = 4,000,000
    const int out_n  = out_size;          // S*32 = 3,200,000
    const int seg_n  = out_size / 32;     // S = 100,000
    const int nquads = rows * 8;          // 32,000,000 eighth-rows (4-float quads)

    // 1) Zero accumulators + counts (d_out/d_ws are poisoned, not re-zeroed
    //    between replays; re-zeroing every call keeps launches deterministic).
    zero_init_kernel<<<(out_n + TPB - 1) / TPB, TPB, 0, stream>>>(
        out, cnt, out_n, seg_n);

    // 2) Streamed scatter-add into L2-resident accumulators.
    scatter_add_kernel<<<(nquads + TPB - 1) / TPB, TPB, 0, stream>>>(
        x, index, out, cnt, nquads, seg_n);

    // 3) Divide by clamped counts.
    finalize_kernel<<<(out_n + TPB - 1) / TPB, TPB, 0, stream>>>(out, cnt, out_n);
}